// FlattenMambaEncoder_13357348290697
// MI455X (gfx1250) — compile-verified
//
#include <hip/hip_runtime.h>
#include <hip/hip_bf16.h>
#include <math.h>

// Sizes from the reference
#define Bb 8
#define Pp 2048
#define Dd 128
#define DIi 256
#define Nn 16
#define Rr 8
#define ROWS (Bb * Pp) // 16384

// Chunked-scan decomposition: 2048 = CH chunks x CL steps
#define CH 32
#define CHLOG 5
#define CL (Pp / CH)   // 64

typedef __attribute__((ext_vector_type(2))) float v2f;
typedef __attribute__((ext_vector_type(8))) float v8f;

__device__ __forceinline__ float silu_f(float x) { return x / (1.0f + __expf(-x)); }
__device__ __forceinline__ float softplus_f(float x) {
    return (x > 20.0f) ? x : __logf(1.0f + __expf(x));
}

// ---------------------------------------------------------------------------
// Generic fp32 WMMA GEMM: C = act(A[M,K] @ W[K,N] + bias) (+ residual)
// Block: 256 threads = 8 waves; block tile = 16 rows x 128 cols, 16 cols/wave.
// V_WMMA_F32_16X16X4_F32 (exact fp32 on the matrix pipe).
// A-frag (ISA 7.12.2, 32-bit A 16x4): v[0]={K0|K2}, v[1]={K1|K3} per half-wave.
// C/D: VGPR i -> rows (i | i+8) across half-waves; column n0+(lane&15) per lane.
// Out-of-range columns: clamp the load index (in-bounds garbage stays in its
// own lane/column) and skip the store — no EXEC masking in the K-loop.
// ---------------------------------------------------------------------------
template <int ACT, bool HAS_BIAS, bool HAS_RES>
__global__ __launch_bounds__(256) void gemm_wmma_f32(
    const float* __restrict__ A, const float* __restrict__ W,
    const float* __restrict__ bias, const float* __restrict__ res,
    float* __restrict__ C, int N, int K, int lda, int ldw, int ldc)
{
    const int lane = threadIdx.x & 31;
    const int wv   = threadIdx.x >> 5;
    const int half = lane >> 4;     // 0: lanes 0-15, 1: lanes 16-31
    const int m    = lane & 15;
    const int row0 = blockIdx.x * 16;
    const int n    = blockIdx.y * 128 + wv * 16 + m;
    const bool nok = (n < N);
    const int nsafe = nok ? n : (N - 1);   // clamped: loads always in-bounds

    v8f acc = {};
    const float*  ap  = A + (size_t)(row0 + m) * lda + 2 * half;
    const float*  wp0 = W + (size_t)(2 * half) * ldw + nsafe;   // row ka
    const float*  wp1 = wp0 + ldw;                              // row ka+1
    const size_t  wstep = (size_t)4 * ldw;

    for (int k = 0; k < K; k += 4) {
        v2f a;
        {   // contiguous K pair -> 8B load (lda even in all uses)
            const float2 t = *(const float2*)ap;
            a.x = t.x; a.y = t.y;
        }
        v2f b;
        b.x = *wp0;
        b.y = *wp1;
        acc = __builtin_amdgcn_wmma_f32_16x16x4_f32(
            /*neg_a=*/false, a, /*neg_b=*/false, b,
            /*c_mod=*/(short)0, acc, /*reuse_a=*/false, /*reuse_b=*/false);
        ap  += 4;
        wp0 += wstep;
        wp1 += wstep;
    }

    if (!nok) return;
    const float bn = HAS_BIAS ? bias[n] : 0.0f;
#pragma unroll
    for (int i = 0; i < 8; ++i) {
        const int r = row0 + i + 8 * half;
        float v = acc[i] + bn;
        if (ACT == 1) v = softplus_f(v);
        if (HAS_RES) v += res[(size_t)r * ldc + n];
        C[(size_t)r * ldc + n] = v;
    }
}

// ---------------------------------------------------------------------------
// Causal depthwise conv (K=3) + SiLU over the u half of xz.
// ---------------------------------------------------------------------------
__global__ __launch_bounds__(256) void conv_silu_kernel(
    const float* __restrict__ xz, const float* __restrict__ convw,
    const float* __restrict__ convb, float* __restrict__ u)
{
    const int idx = blockIdx.x * 256 + threadIdx.x;   // ROWS*DI threads
    const int d   = idx & (DIi - 1);
    const int row = idx >> 8;       // b*P + p
    const int p   = row & (Pp - 1);

    const float w0 = convw[d * 3 + 0];
    const float w1 = convw[d * 3 + 1];
    const float w2 = convw[d * 3 + 2];

    float v = xz[(size_t)row * (2 * DIi) + d] * w2;
    if (p >= 1) v += xz[(size_t)(row - 1) * (2 * DIi) + d] * w1;
    if (p >= 2) v += xz[(size_t)(row - 2) * (2 * DIi) + d] * w0;
    v += convb[d];
    u[idx] = silu_f(v);
}

// ---------------------------------------------------------------------------
// Chunked selective scan. Lane layout: t = ((b*DI+d)*CH + c)*16 + n, so the
// 16 N-lanes of a (b,d,chunk) group are contiguous and 16-aligned (wave32 ->
// two groups per wave; shfl_xor width 16 reduces over N).
//
// Pass 1: per chunk, local scan from h=0; record h_end and decay product.
// Pass 2: 32-step sequential combine per (b,d,n) -> carry-in per chunk.
// Pass 3: rerun chunk from its true carry-in; fused u*Dp skip + SiLU(z) gate.
// dA = exp(dt*A) is in (0,1] (dt>=0, A<0), so recombination is stable.
// ---------------------------------------------------------------------------
__global__ __launch_bounds__(256) void scan_pass1_kernel(
    const float* __restrict__ dt, const float* __restrict__ u,
    const float* __restrict__ dbl, const float* __restrict__ Alog,
    float* __restrict__ hend, float* __restrict__ pprd)
{
    const int t  = blockIdx.x * 256 + threadIdx.x;
    const int n  = t & (Nn - 1);
    const int g  = t >> 4;              // (b*DI + d)*CH + c
    const int c  = g & (CH - 1);
    const int bd = g >> CHLOG;
    const int d  = bd & (DIi - 1);
    const int b  = bd >> 8;

    const float A = -__expf(Alog[d * Nn + n]);
    size_t row = (size_t)b * Pp + (size_t)c * CL;

    float h = 0.0f, pr = 1.0f;
    for (int i = 0; i < CL; ++i, ++row) {
        const float dtv = dt[row * DIi + d];
        const float uv  = u[row * DIi + d];
        const float Bv  = dbl[row * (Rr + 2 * Nn) + Rr + n];
        const float dA  = __expf(dtv * A);
        h  = dA * h + (dtv * uv) * Bv;
        pr *= dA;
    }
    hend[t] = h;
    pprd[t] = pr;
}

__global__ __launch_bounds__(256) void scan_pass2_kernel(
    const float* __restrict__ hend, const float* __restrict__ pprd,
    float* __restrict__ h0)
{
    const int t  = blockIdx.x * 256 + threadIdx.x;   // B*DI*N = 32768 lanes
    const int n  = t & (Nn - 1);
    const int bd = t >> 4;
    size_t idx = (size_t)bd * CH * Nn + n;

    float carry = 0.0f;
    for (int c = 0; c < CH; ++c, idx += Nn) {
        h0[idx] = carry;
        carry = pprd[idx] * carry + hend[idx];
    }
}

__global__ __launch_bounds__(256) void scan_pass3_kernel(
    const float* __restrict__ dt, const float* __restrict__ u,
    const float* __restrict__ dbl, const float* __restrict__ xz,
    const float* __restrict__ Alog, const float* __restrict__ Dp,
    const float* __restrict__ h0, float* __restrict__ y)
{
    const int t  = blockIdx.x * 256 + threadIdx.x;
    const int n  = t & (Nn - 1);
    const int g  = t >> 4;
    const int c  = g & (CH - 1);
    const int bd = g >> CHLOG;
    const int d  = bd & (DIi - 1);
    const int b  = bd >> 8;

    const float A   = -__expf(Alog[d * Nn + n]);
    const float dpd = Dp[d];
    size_t row = (size_t)b * Pp + (size_t)c * CL;

    float h = h0[t];
    for (int i = 0; i < CL; ++i, ++row) {
        const float dtv = dt[row * DIi + d];
        const float uv  = u[row * DIi + d];
        const float Bv  = dbl[row * (Rr + 2 * Nn) + Rr + n];
        const float Cv  = dbl[row * (Rr + 2 * Nn) + Rr + Nn + n];

        const float dA = __expf(dtv * A);
        h = dA * h + (dtv * uv) * Bv;

        float part = h * Cv;
        part += __shfl_xor(part, 1, 16);
        part += __shfl_xor(part, 2, 16);
        part += __shfl_xor(part, 4, 16);
        part += __shfl_xor(part, 8, 16);

        if (n == 0) {
            const float z = xz[row * (2 * DIi) + DIi + d];
            y[row * DIi + d] = (part + uv * dpd) * silu_f(z);
        }
    }
}

// ---------------------------------------------------------------------------
// LayerNorm over D=128: one wave per row, 4 elements per lane (float4).
// ---------------------------------------------------------------------------
__global__ __launch_bounds__(256) void layernorm128_kernel(
    const float* __restrict__ x, const float* __restrict__ gamma,
    const float* __restrict__ beta, float* __restrict__ out)
{
    const int lane = threadIdx.x & 31;
    const int wave = (blockIdx.x * 256 + threadIdx.x) >> 5;   // row id
    const size_t base = (size_t)wave * Dd + lane * 4;

    float4 v = *(const float4*)(x + base);
    float s = v.x + v.y + v.z + v.w;
    float q = v.x * v.x + v.y * v.y + v.z * v.z + v.w * v.w;
#pragma unroll
    for (int off = 16; off; off >>= 1) {
        s += __shfl_xor(s, off, 32);
        q += __shfl_xor(q, off, 32);
    }
    const float mu  = s * (1.0f / Dd);
    const float var = q * (1.0f / Dd) - mu * mu;
    const float inv = rsqrtf(var + 1e-5f);

    const float4 g  = *(const float4*)(gamma + lane * 4);
    const float4 be = *(const float4*)(beta + lane * 4);
    float4 o;
    o.x = (v.x - mu) * inv * g.x + be.x;
    o.y = (v.y - mu) * inv * g.y + be.y;
    o.z = (v.z - mu) * inv * g.z + be.z;
    o.w = (v.w - mu) * inv * g.w + be.w;
    *(float4*)(out + base) = o;
}

// ---------------------------------------------------------------------------
extern "C" void kernel_launch(void* const* d_in, const int* in_sizes, int n_in,
                              void* d_out, int out_size, void* d_ws, size_t ws_size,
                              hipStream_t stream)
{
    const float* rp    = (const float*)d_in[0];   // (8,2048,32,32)
    const float* Wp    = (const float*)d_in[1];   // (1024,128)
    const float* bp    = (const float*)d_in[2];   // (128)
    const float* Win   = (const float*)d_in[3];   // (2,128,512)
    const float* b_in  = (const float*)d_in[4];   // (2,512)
    const float* convw = (const float*)d_in[5];   // (2,256,3)
    const float* convb = (const float*)d_in[6];   // (2,256)
    const float* Wx    = (const float*)d_in[7];   // (2,256,40)
    const float* Wdt   = (const float*)d_in[8];   // (2,8,256)
    const float* bdt   = (const float*)d_in[9];   // (2,256)
    const float* Alog  = (const float*)d_in[10];  // (2,256,16)
    const float* Dp    = (const float*)d_in[11];  // (2,256)
    const float* Wout  = (const float*)d_in[12];  // (2,256,128)
    const float* bout  = (const float*)d_in[13];  // (2,128)
    const float* gamma = (const float*)d_in[14];  // (128)
    const float* beta  = (const float*)d_in[15];  // (128)

    // Workspace layout (fp32): ~108 MiB
    float* x    = (float*)d_ws;                          // 16384*128
    float* xz   = x    + (size_t)ROWS * Dd;              // 16384*512
    float* u    = xz   + (size_t)ROWS * 2 * DIi;         // 16384*256
    float* dbl  = u    + (size_t)ROWS * DIi;             // 16384*40
    float* dt   = dbl  + (size_t)ROWS * (Rr + 2 * Nn);   // 16384*256
    float* y    = dt   + (size_t)ROWS * DIi;             // 16384*256
    float* hend = y    + (size_t)ROWS * DIi;             // 8*256*16*CH
    float* pprd = hend + (size_t)Bb * DIi * Nn * CH;
    float* h0   = pprd + (size_t)Bb * DIi * Nn * CH;

    const dim3 blk(256);
    const int MB = ROWS / 16;                     // 1024 row-tiles
    const int SCAN_BLKS = (Bb * DIi * Nn * CH) / 256;   // 4096

    // x = rp.reshape(.,1024) @ Wp + bp       [M=16384, K=1024, N=128]
    gemm_wmma_f32<0, true, false><<<dim3(MB, 1), blk, 0, stream>>>(
        rp, Wp, bp, nullptr, x, Dd, 1024, 1024, Dd, Dd);

    for (int l = 0; l < 2; ++l) {
        const float* Win_l   = Win   + (size_t)l * Dd * 2 * DIi;
        const float* b_in_l  = b_in  + (size_t)l * 2 * DIi;
        const float* convw_l = convw + (size_t)l * DIi * 3;
        const float* convb_l = convb + (size_t)l * DIi;
        const float* Wx_l    = Wx    + (size_t)l * DIi * (Rr + 2 * Nn);
        const float* Wdt_l   = Wdt   + (size_t)l * Rr * DIi;
        const float* bdt_l   = bdt   + (size_t)l * DIi;
        const float* Alog_l  = Alog  + (size_t)l * DIi * Nn;
        const float* Dp_l    = Dp    + (size_t)l * DIi;
        const float* Wout_l  = Wout  + (size_t)l * DIi * Dd;
        const float* bout_l  = bout  + (size_t)l * Dd;

        // xz = x @ Win + b_in                [K=128, N=512]
        gemm_wmma_f32<0, true, false><<<dim3(MB, 4), blk, 0, stream>>>(
            x, Win_l, b_in_l, nullptr, xz, 2 * DIi, Dd, Dd, 2 * DIi, 2 * DIi);

        // u = silu(causal_conv(xz[:, :256]))
        conv_silu_kernel<<<(ROWS * DIi) / 256, blk, 0, stream>>>(
            xz, convw_l, convb_l, u);

        // dbl = u @ Wx                       [K=256, N=40]
        gemm_wmma_f32<0, false, false><<<dim3(MB, 1), blk, 0, stream>>>(
            u, Wx_l, nullptr, nullptr, dbl, Rr + 2 * Nn, DIi, DIi,
            Rr + 2 * Nn, Rr + 2 * Nn);

        // dt = softplus(dbl[:, :8] @ Wdt + bdt)   [K=8, N=256]
        gemm_wmma_f32<1, true, false><<<dim3(MB, 2), blk, 0, stream>>>(
            dbl, Wdt_l, bdt_l, nullptr, dt, DIi, Rr, Rr + 2 * Nn, DIi, DIi);

        // chunked selective scan + gating -> y
        scan_pass1_kernel<<<SCAN_BLKS, blk, 0, stream>>>(
            dt, u, dbl, Alog_l, hend, pprd);
        scan_pass2_kernel<<<(Bb * DIi * Nn) / 256, blk, 0, stream>>>(
            hend, pprd, h0);
        scan_pass3_kernel<<<SCAN_BLKS, blk, 0, stream>>>(
            dt, u, dbl, xz, Alog_l, Dp_l, h0, y);

        // x = x + y @ Wout + bout            [K=256, N=128]
        gemm_wmma_f32<0, true, true><<<dim3(MB, 1), blk, 0, stream>>>(
            y, Wout_l, bout_l, x, x, Dd, DIi, DIi, Dd, Dd);
    }

    // final LayerNorm -> d_out
    layernorm128_kernel<<<ROWS / 8, blk, 0, stream>>>(
        x, gamma, beta, (float*)d_out);
}